// Codebook_59107339928241
// MI455X (gfx1250) — compile-verified
//
#include <hip/hip_runtime.h>
#include <hip/hip_bf16.h>

typedef __attribute__((ext_vector_type(2))) float v2f;
typedef __attribute__((ext_vector_type(8))) float v8f;
typedef __attribute__((ext_vector_type(4))) int   v4i;

typedef __attribute__((address_space(1))) v4i* gptr_v4i;   // global int4*
typedef __attribute__((address_space(3))) v4i* lptr_v4i;   // LDS int4*

#define DIM 256
#define NCODES 1024
#define NTOKENS (16 * 1024)
#define ROWPAD 260          // 256 + 4 floats pad -> conflict-free ds_load_b64
#define NTILES (NCODES / 16)

// ---------------------------------------------------------------------------
// CDNA5 async copy helpers (guarded; fall back to sync copy if unavailable)
// ---------------------------------------------------------------------------
#ifndef __has_builtin
#define __has_builtin(x) 0
#endif

__device__ __forceinline__ void async_wait0() {
#if __has_builtin(__builtin_amdgcn_s_wait_asynccnt)
    __builtin_amdgcn_s_wait_asynccnt(0);
#else
    asm volatile("s_wait_asynccnt 0x0" ::: "memory");
#endif
}

#if __has_builtin(__builtin_amdgcn_global_load_async_to_lds_b128)
#define HAVE_ASYNC_LDS 1
__device__ __forceinline__ void async_cp16(const float* g, float* l) {
    __builtin_amdgcn_global_load_async_to_lds_b128(
        (gptr_v4i)(g), (lptr_v4i)(l), 0, 0);
}
#else
#define HAVE_ASYNC_LDS 0
__device__ __forceinline__ void async_cp16(const float* g, float* l) {
    *(float4*)l = *(const float4*)g;   // sync fallback: global->reg->LDS
}
#endif

// ---------------------------------------------------------------------------
// Kernel 1: e_sq[n] = sum_d E[n][d]^2   (one wave per code row)
// ---------------------------------------------------------------------------
__global__ void vq_esq_kernel(const float* __restrict__ e, float* __restrict__ esq) {
    const int wave = (blockIdx.x * blockDim.x + threadIdx.x) >> 5;
    const int lane = threadIdx.x & 31;
    if (wave >= NCODES) return;
    const float* row = e + (size_t)wave * DIM;
    float s = 0.f;
#pragma unroll
    for (int i = 0; i < DIM / 32; ++i) {
        float v = row[lane + i * 32];
        s = fmaf(v, v, s);
    }
#pragma unroll
    for (int m = 16; m >= 1; m >>= 1)
        s += __shfl_xor(s, m, 32);
    if (lane == 0) esq[wave] = s;
}

// ---------------------------------------------------------------------------
// Kernel 2: fused  score = e_sq - 2*x.e  via V_WMMA_F32_16X16X4_F32.
// B (E tiles) staged in LDS via async-to-LDS copies, double buffered;
// one wave per 16-token tile, A resident in 128 VGPRs.
// ---------------------------------------------------------------------------
__global__ void __launch_bounds__(256)
vq_argmin_gather_kernel(const float* __restrict__ x,
                        const float* __restrict__ e,
                        const float* __restrict__ esq,
                        float* __restrict__ out) {
    __shared__ float smem[2][16 * ROWPAD];    // 2 x 16.6 KB double buffer

    const int tid    = threadIdx.x;
    const int lane   = tid & 31;
    const int waveId = tid >> 5;
    const int tile   = blockIdx.x * (blockDim.x >> 5) + waveId;
    const int token0 = tile * 16;
    const int lm     = lane & 15;       // row (A) / col (B) within tile
    const int kOff   = (lane >> 4) * 2; // upper half lanes hold K+2,K+3

    // Cooperative copy of code tile ct (16 rows x 256 f32 = 16 KB) into smem[buf].
    // 256 threads: thread t -> row t/16, 4 x b128 chunks strided by 64 floats.
    const int cpRow = tid >> 4;
    const int cpCol = (tid & 15) * 4;
    auto issue_copy = [&](int ct, int buf) {
        const float* src = e + (size_t)(ct * 16 + cpRow) * DIM + cpCol;
        float* dst = &smem[buf][cpRow * ROWPAD + cpCol];
#pragma unroll
        for (int i = 0; i < 4; ++i)
            async_cp16(src + i * 64, dst + i * 64);
    };

    // ---- Load A tile once: WMMA f32 16x16x4 A layout (v2f per K-step).
    v2f a[64];
    const float* arow = x + (size_t)(token0 + lm) * DIM + kOff;
#pragma unroll
    for (int kk = 0; kk < 64; ++kk)
        a[kk] = *(const v2f*)(arow + kk * 4);

    float best[8];
    int   bidx[8];
#pragma unroll
    for (int j = 0; j < 8; ++j) { best[j] = 3.4e38f; bidx[j] = 0; }

    // ---- Prologue: stage tile 0.
    issue_copy(0, 0);
    async_wait0();
    __syncthreads();

    int buf = 0;
    for (int ct = 0; ct < NTILES; ++ct) {
        if (ct + 1 < NTILES) issue_copy(ct + 1, buf ^ 1);   // prefetch next tile

        const float* brow = &smem[buf][lm * ROWPAD + kOff]; // B[k][n] = E[n][k]
        v8f acc = {};
#pragma unroll
        for (int kk = 0; kk < 64; ++kk) {
            v2f b = *(const v2f*)(brow + kk * 4);            // ds_load_b64
            acc = __builtin_amdgcn_wmma_f32_16x16x4_f32(
                false, a[kk], false, b, (short)0, acc, false, false);
        }

        // C/D layout: component j -> row (lane<16 ? j : j+8), col ct*16+lm
        const int n = ct * 16 + lm;
        const float es = esq[n];
#pragma unroll
        for (int j = 0; j < 8; ++j) {
            float sc = fmaf(-2.f, acc[j], es);
            if (sc < best[j]) { best[j] = sc; bidx[j] = n; }
        }

        async_wait0();      // next tile landed in LDS
        __syncthreads();    // everyone done reading buf before it is reused
        buf ^= 1;
    }

    // ---- Min-reduce (value, index) across the 16 lanes of each half.
#pragma unroll
    for (int j = 0; j < 8; ++j) {
#pragma unroll
        for (int m = 8; m >= 1; m >>= 1) {
            float ov = __shfl_xor(best[j], m, 32);
            int   oi = __shfl_xor(bidx[j], m, 32);
            if (ov < best[j] || (ov == best[j] && oi < bidx[j])) {
                best[j] = ov; bidx[j] = oi;
            }
        }
    }

    // ---- Gather: out[t] = (x[t] + E[idx_t]) - x[t], 256 floats per token.
#pragma unroll
    for (int t = 0; t < 16; ++t) {
        const int idx = __shfl(bidx[t & 7], (t < 8) ? 0 : 16, 32);
        const float4* q  = (const float4*)(e   + (size_t)idx * DIM);
        const float4* xv = (const float4*)(x   + (size_t)(token0 + t) * DIM);
        float4*       o  = (float4*)      (out + (size_t)(token0 + t) * DIM);
#pragma unroll
        for (int c = 0; c < 2; ++c) {
            const int off = c * 32 + lane;   // 64 float4 per 256-float row
            float4 qq = q[off];
            float4 xx = xv[off];
            float4 r;
            r.x = (xx.x + qq.x) - xx.x;
            r.y = (xx.y + qq.y) - xx.y;
            r.z = (xx.z + qq.z) - xx.z;
            r.w = (xx.w + qq.w) - xx.w;
            o[off] = r;
        }
    }
}

// ---------------------------------------------------------------------------
extern "C" void kernel_launch(void* const* d_in, const int* in_sizes, int n_in,
                              void* d_out, int out_size, void* d_ws, size_t ws_size,
                              hipStream_t stream) {
    const float* x = (const float*)d_in[0];          // [16,1024,256] f32
    const float* e = (const float*)d_in[1];          // [1024,256]    f32
    float* out = (float*)d_out;                      // [16,1024,256] f32
    float* esq = (float*)d_ws;                       // [1024] f32 scratch

    vq_esq_kernel<<<(NCODES * 32) / 256, 256, 0, stream>>>(e, esq);
    vq_argmin_gather_kernel<<<NTOKENS / 128, 256, 0, stream>>>(x, e, esq, out);
}